// GCN_50190987821616
// MI455X (gfx1250) — compile-verified
//
#include <hip/hip_runtime.h>

#define N_NODES 100000
#define N_EDGES 3200000
#define NFEAT   512
#define NHID    256
#define NCLS    64

typedef __attribute__((ext_vector_type(2))) float v2f;
typedef __attribute__((ext_vector_type(8))) float v8f;

// ---------------------------------------------------------------------------
// Fused GEMM + bias:  out[M,NOUT] = A[M,K] @ W[K,NOUT] + bias
// One wave computes a 16x16 tile with V_WMMA_F32_16X16X4_F32 (fp32-exact).
//
// Operand layouts (CDNA5 ISA 7.12.2, wave32):
//  A 16x4 fp32 : lanes 0-15 -> M=lane,   VGPR0=K0, VGPR1=K1
//                lanes 16-31 -> M=lane-16, VGPR0=K2, VGPR1=K3
//  B 4x16 fp32 : mirrored (lanes 0-15 -> N, K{0,1}; lanes 16-31 -> N, K{2,3})
//  C/D 16x16   : VGPR r, lanes 0-15 -> M=r, N=lane; lanes 16-31 -> M=r+8
// ---------------------------------------------------------------------------
template <int K, int NOUT, int WAVES>
__global__ __launch_bounds__(WAVES * 32) void gemm_bias_wmma(
    const float* __restrict__ A, const float* __restrict__ W,
    const float* __restrict__ bias, float* __restrict__ out)
{
    const int lane = threadIdx.x & 31;
    const int wave = threadIdx.x >> 5;
    const int half = lane >> 4;   // 0: K{0,1} / M row group, 1: K{2,3}
    const int l16  = lane & 15;

    const int rowBase = blockIdx.x * 16;                     // M tile
    const int nBase   = (blockIdx.y * WAVES + wave) * 16;    // N tile

    const float* aRow = A + (size_t)(rowBase + l16) * K + 2 * half;
    const float* wCol = W + (size_t)(2 * half) * NOUT + nBase + l16;

    v8f acc = {};
#pragma unroll 4
    for (int k = 0; k < K; k += 4) {
        if ((k & 63) == 0)  // stride ahead on the fp32 A row (global_prefetch_b8)
            __builtin_prefetch(aRow + k + 64, 0, 0);

        v2f av;
        {
            const float2 t = *(const float2*)(aRow + k);
            av.x = t.x; av.y = t.y;
        }
        v2f bv;
        bv.x = wCol[(size_t)k * NOUT];
        bv.y = wCol[(size_t)k * NOUT + NOUT];

        acc = __builtin_amdgcn_wmma_f32_16x16x4_f32(
            /*neg_a=*/false, av, /*neg_b=*/false, bv,
            /*c_mod=*/(short)0, acc, /*reuse_a=*/false, /*reuse_b=*/false);
    }

    const int   n  = nBase + l16;
    const float bn = bias[n];
#pragma unroll
    for (int r = 0; r < 8; ++r) {
        const int m = rowBase + r + 8 * half;
        out[(size_t)m * NOUT + n] = acc[r] + bn;
    }
}

// ---------------------------------------------------------------------------
// COO SpMM scatter:  out[rows[e], :] += vals[e] * H[cols[e], :]
// D/4 threads per edge, consecutive threads -> consecutive float4 of one edge
// (coalesced b128 gather; z0 [102 MB] fits in the 192 MB L2, so the random
// row gathers should be L2-resident). Scatter via global_atomic_add_f32.
// ---------------------------------------------------------------------------
template <int D>
__global__ __launch_bounds__(256) void spmm_scatter(
    const int* __restrict__ rows, const int* __restrict__ cols,
    const float* __restrict__ vals, const float* __restrict__ H,
    float* __restrict__ out, int nEdges)
{
    constexpr int CH = D / 4;  // float4 chunks per edge (power of two)
    const long long gid = (long long)blockIdx.x * blockDim.x + threadIdx.x;
    if (gid >= (long long)nEdges * CH) return;

    const int e = (int)(gid / CH);
    const int c = (int)(gid % CH);

    const int   src = cols[e];
    const int   dst = rows[e];
    const float v   = vals[e];

    const float4 hv = *(const float4*)(H + (size_t)src * D + c * 4);
    float* o = out + (size_t)dst * D + c * 4;
    atomicAdd(o + 0, v * hv.x);
    atomicAdd(o + 1, v * hv.y);
    atomicAdd(o + 2, v * hv.z);
    atomicAdd(o + 3, v * hv.w);
}

// In-place ReLU over float4 chunks.
__global__ __launch_bounds__(256) void relu4(float* __restrict__ p, long long n4)
{
    const long long i = (long long)blockIdx.x * blockDim.x + threadIdx.x;
    if (i >= n4) return;
    float4 t = ((const float4*)p)[i];
    t.x = fmaxf(t.x, 0.0f);
    t.y = fmaxf(t.y, 0.0f);
    t.z = fmaxf(t.z, 0.0f);
    t.w = fmaxf(t.w, 0.0f);
    ((float4*)p)[i] = t;
}

extern "C" void kernel_launch(void* const* d_in, const int* in_sizes, int n_in,
                              void* d_out, int out_size, void* d_ws, size_t ws_size,
                              hipStream_t stream)
{
    (void)in_sizes; (void)n_in; (void)ws_size;

    const float* x     = (const float*)d_in[0];
    const int*   rows0 = (const int*)  d_in[1];
    const int*   cols0 = (const int*)  d_in[2];
    const float* vals0 = (const float*)d_in[3];
    const int*   rows1 = (const int*)  d_in[4];
    const int*   cols1 = (const int*)  d_in[5];
    const float* vals1 = (const float*)d_in[6];
    const float* W0    = (const float*)d_in[7];
    const float* b0    = (const float*)d_in[8];
    const float* W1    = (const float*)d_in[9];
    const float* b1    = (const float*)d_in[10];
    float*       out   = (float*)d_out;

    // Workspace: z0 [N,256] then agg0 [N,256].  z1 [N,64] reuses z0.
    float* z0   = (float*)d_ws;
    float* agg0 = z0 + (size_t)N_NODES * NHID;

    // ---- Layer 0: linear (WMMA) -> scatter-add -> relu ----
    {
        dim3 grid(N_NODES / 16, NHID / (16 * 4));  // 6250 x 4, 4 waves/block
        gemm_bias_wmma<NFEAT, NHID, 4><<<grid, 128, 0, stream>>>(x, W0, b0, z0);
    }
    hipMemsetAsync(agg0, 0, (size_t)N_NODES * NHID * sizeof(float), stream);
    {
        const long long work = (long long)N_EDGES * (NHID / 4);
        spmm_scatter<NHID><<<(unsigned)((work + 255) / 256), 256, 0, stream>>>(
            rows0, cols0, vals0, z0, agg0, N_EDGES);
    }
    {
        const long long n4 = (long long)N_NODES * NHID / 4;
        relu4<<<(unsigned)((n4 + 255) / 256), 256, 0, stream>>>(agg0, n4);
    }

    // ---- Layer 1: linear (WMMA) -> scatter-add -> relu ----
    {
        dim3 grid(N_NODES / 16, NCLS / (16 * 4));  // 6250 x 1
        gemm_bias_wmma<NHID, NCLS, 4><<<grid, 128, 0, stream>>>(agg0, W1, b1, z0);
    }
    hipMemsetAsync(out, 0, (size_t)out_size * sizeof(float), stream);
    {
        const long long work = (long long)N_EDGES * (NCLS / 4);
        spmm_scatter<NCLS><<<(unsigned)((work + 255) / 256), 256, 0, stream>>>(
            rows1, cols1, vals1, z0, out, N_EDGES);
    }
    {
        const long long n4 = (long long)N_NODES * NCLS / 4;
        relu4<<<(unsigned)((n4 + 255) / 256), 256, 0, stream>>>(out, n4);
    }
}